// KeypointLoss_51866025066626
// MI455X (gfx1250) — compile-verified
//
#include <hip/hip_runtime.h>

typedef __attribute__((ext_vector_type(2))) float v2f;
typedef __attribute__((ext_vector_type(4))) float v4f;
typedef __attribute__((ext_vector_type(8))) float v8f;

#define SDIM 5
#define BDIM 16
#define CDIM 11
#define HW   16384              // 128*128, contiguous per (s,b,c)
#define NBLK (SDIM * BDIM * CDIM)   // 880 blocks
#define TPB  256                // 8 waves of 32
#define V4T  (HW / TPB / 4)     // 16 float4 loads per thread per array

// Cross-lane sum of `acc` over all 32 lanes using V_WMMA_F32_16X16X4_F32.
// A (16x4): VGPR0 = per-lane partials (32 distinct entries), VGPR1 = 0.
// B (4x16): all ones.  D[m][n] = rowsum(A[m]) replicated over N, so
// sum(d[0..7]) gives rows 0-7 in lanes 0-15 and rows 8-15 in lanes 16-31;
// one shfl_xor(16) completes the total in every lane.
__device__ __forceinline__ float wave_sum_wmma(float acc) {
    v2f a; a[0] = acc;  a[1] = 0.0f;
    v2f b; b[0] = 1.0f; b[1] = 1.0f;
    v8f c = {};
    v8f d = __builtin_amdgcn_wmma_f32_16x16x4_f32(
        /*neg_a=*/false, a, /*neg_b=*/false, b,
        /*c_mod=*/(short)0, c, /*reuse_a=*/false, /*reuse_b=*/false);
    float r = ((d[0] + d[1]) + (d[2] + d[3])) + ((d[4] + d[5]) + (d[6] + d[7]));
    r += __shfl_xor(r, 16, 32);
    return r;
}

// One block per (s,b,c) slice: stream 2*64KB with coalesced float4 NT loads,
// accumulate diff^2 in f32, reduce wave->block, write one partial to ws.
__global__ void __launch_bounds__(TPB)
heat_partial_kernel(const float* __restrict__ hp,
                    const float* __restrict__ hm,
                    float* __restrict__ ws) {
    const int blk = blockIdx.x;
    const size_t base = (size_t)blk * HW;
    const v4f* p4 = (const v4f*)(hp + base);
    const v4f* m4 = (const v4f*)(hm + base);

    float acc = 0.0f;
    int idx = threadIdx.x;
#pragma unroll
    for (int t = 0; t < V4T; ++t, idx += TPB) {
        v4f p = __builtin_nontemporal_load(p4 + idx);
        v4f m = __builtin_nontemporal_load(m4 + idx);
        float d0 = p[0] - m[0];
        float d1 = p[1] - m[1];
        float d2 = p[2] - m[2];
        float d3 = p[3] - m[3];
        acc = fmaf(d0, d0, acc);
        acc = fmaf(d1, d1, acc);
        acc = fmaf(d2, d2, acc);
        acc = fmaf(d3, d3, acc);
    }

    // all 256 threads reach here with full EXEC -> WMMA legal
    float wsum = wave_sum_wmma(acc);

    __shared__ float red[TPB / 32];
    const int wave = threadIdx.x >> 5;
    if ((threadIdx.x & 31) == 0) red[wave] = wsum;
    __syncthreads();
    if (threadIdx.x == 0) {
        float s = 0.0f;
#pragma unroll
        for (int i = 0; i < TPB / 32; ++i) s += red[i];
        ws[blk] = s;   // fixed-order, deterministic
    }
}

// 80 outputs: fold the C partials per (b,s) with the 1/HW mean, and compute
// the tiny 77-element label loss. Fixed summation order -> deterministic.
__global__ void finalize_kernel(const float* __restrict__ ws,
                                const float* __restrict__ lp,
                                const float* __restrict__ lab,
                                float* __restrict__ out) {
    const int t = threadIdx.x;
    if (t >= SDIM * BDIM) return;
    const int b = t / SDIM;
    const int s = t % SDIM;

    // combined_loss[b, s]
    float hsum = 0.0f;
    const float* w = ws + ((size_t)s * BDIM + b) * CDIM;
#pragma unroll
    for (int c = 0; c < CDIM; ++c) hsum += w[c];
    out[b * SDIM + s] = hsum * (1.0f / (float)HW);

    // labels_loss[b, s]: label_preds[s,b,:,:7] vs labels[b,:,:7]; F==7 so the
    // 11*7 = 77 floats are contiguous in both tensors.
    float lsum = 0.0f;
    const float* lps = lp + (size_t)(s * BDIM + b) * CDIM * 7;
    const float* lbs = lab + (size_t)b * CDIM * 7;
#pragma unroll
    for (int i = 0; i < CDIM * 7; ++i) {
        float d = lps[i] - lbs[i];
        lsum = fmaf(d, d, lsum);
    }
    out[SDIM * BDIM + b * SDIM + s] = lsum;
}

extern "C" void kernel_launch(void* const* d_in, const int* in_sizes, int n_in,
                              void* d_out, int out_size, void* d_ws, size_t ws_size,
                              hipStream_t stream) {
    const float* heat_preds  = (const float*)d_in[0];
    const float* heatmaps    = (const float*)d_in[1];
    const float* label_preds = (const float*)d_in[2];
    const float* labels      = (const float*)d_in[3];
    float* out = (float*)d_out;
    float* ws  = (float*)d_ws;   // needs 880 floats = 3.5 KB

    heat_partial_kernel<<<NBLK, TPB, 0, stream>>>(heat_preds, heatmaps, ws);
    finalize_kernel<<<1, 128, 0, stream>>>(ws, label_preds, labels, out);
}